// LightGCN_49624052138449
// MI455X (gfx1250) — compile-verified
//
#include <hip/hip_runtime.h>

#define NUM_USERS 100000
#define NUM_ITEMS 50000
#define N_NODES   150000
#define LATENT    64
#define N_EDGES   4000000    // 2 * N_INTERACTIONS
#define N_QUERIES 16384
#define N_LAYERS  3

typedef __attribute__((ext_vector_type(2))) float v2f;
typedef __attribute__((ext_vector_type(8))) float v8f;

// Guaranteed-native f32 atomic add (no CAS expansion): no-return form,
// device scope so the RMW executes at L2 (which owns the atomic units).
__device__ __forceinline__ void atomic_add_f32_dev(float* p, float v) {
    asm volatile("global_atomic_add_f32 %0, %1, off scope:SCOPE_DEV"
                 :: "v"(p), "v"(v) : "memory");
}

// ---------------------------------------------------------------------------
// init: acc = x = concat(user_emb, item_emb), vectorized float4
// ---------------------------------------------------------------------------
__global__ void lgcn_init(const float4* __restrict__ ue, const float4* __restrict__ ie,
                          float4* __restrict__ acc, float4* __restrict__ x) {
    const int total = N_NODES * (LATENT / 4);           // 2.4M float4
    const int ucut  = NUM_USERS * (LATENT / 4);
    for (int i = blockIdx.x * blockDim.x + threadIdx.x; i < total;
         i += gridDim.x * blockDim.x) {
        float4 v = (i < ucut) ? ue[i] : ie[i - ucut];
        acc[i] = v;
        x[i]   = v;
    }
}

// ---------------------------------------------------------------------------
// zero: y = 0
// ---------------------------------------------------------------------------
__global__ void lgcn_zero(float4* __restrict__ y) {
    const int total = N_NODES * (LATENT / 4);
    float4 z; z.x = 0.f; z.y = 0.f; z.z = 0.f; z.w = 0.f;
    for (int i = blockIdx.x * blockDim.x + threadIdx.x; i < total;
         i += gridDim.x * blockDim.x)
        y[i] = z;
}

// ---------------------------------------------------------------------------
// SpMM: one wave32 per edge. Lane L handles dims [2L, 2L+1]:
//   - edge metadata scalarized via readfirstlane -> s_load through K$
//   - gather = one coalesced 256B row read (L2-resident, 38.4MB << 192MB L2)
//   - scatter = coalesced native global_atomic_add_f32, device scope
// Launched with exactly N_EDGES*32 threads -> no bounds check needed.
// ---------------------------------------------------------------------------
__global__ void lgcn_spmm(const float* __restrict__ vals, const int* __restrict__ rows,
                          const int* __restrict__ cols, const float* __restrict__ x,
                          float* __restrict__ y) {
    unsigned wid  = blockIdx.x * (blockDim.x >> 5) + (threadIdx.x >> 5);
    unsigned e    = (unsigned)__builtin_amdgcn_readfirstlane((int)wid); // wave-uniform
    unsigned lane = threadIdx.x & 31u;

    int   r = rows[e];          // uniform -> scalar loads via constant cache
    int   c = cols[e];
    float v = vals[e];

    const float2 s = *(const float2*)(x + (size_t)c * LATENT + lane * 2u);
    float* dst = y + (size_t)r * LATENT + lane * 2u;
    atomic_add_f32_dev(dst,     v * s.x);
    atomic_add_f32_dev(dst + 1, v * s.y);
}

// ---------------------------------------------------------------------------
// acc += y
// ---------------------------------------------------------------------------
__global__ void lgcn_accadd(float4* __restrict__ acc, const float4* __restrict__ y) {
    const int total = N_NODES * (LATENT / 4);
    for (int i = blockIdx.x * blockDim.x + threadIdx.x; i < total;
         i += gridDim.x * blockDim.x) {
        float4 a = acc[i];
        float4 b = y[i];
        a.x += b.x; a.y += b.y; a.z += b.z; a.w += b.w;
        acc[i] = a;
    }
}

// ---------------------------------------------------------------------------
// Final: gamma[q] = dot(acc[users[q]], acc[NUM_USERS+items[q]]) / 16
// One wave per 16 queries: diag(U x I^T) via 16 chained v_wmma_f32_16x16x4_f32.
// A (16x4 f32): lanes 0-15 -> M=lane, K={0,1}; lanes 16-31 -> M=lane-16, K={2,3}.
// B = I^T (4x16): same gather pattern on I rows by transpose symmetry.
// C/D 16x16 f32: diag (m,m) lives in VGPR m lane m (m<8), VGPR m-8 lane m+16 (m>=8).
// Launched with EXACTLY 1024 waves so EXEC is all ones (WMMA requirement).
// ---------------------------------------------------------------------------
__global__ void lgcn_gamma_wmma(const float* __restrict__ acc,
                                const int* __restrict__ users,
                                const int* __restrict__ items,
                                float* __restrict__ out) {
    unsigned tid  = blockIdx.x * blockDim.x + threadIdx.x;
    unsigned wave = tid >> 5;                 // 0..1023
    unsigned lane = tid & 31u;
    unsigned qrow = lane & 15u;               // M for A, N for B
    unsigned q    = wave * 16u + qrow;

    const float* U = acc + (size_t)users[q] * LATENT;
    const float* I = acc + (size_t)(NUM_USERS + items[q]) * LATENT;
    unsigned koff  = (lane < 16u) ? 0u : 2u;

    v8f c = {};
#pragma unroll
    for (int kk = 0; kk < 16; ++kk) {
        unsigned k = kk * 4u + koff;
        v2f a; a.x = U[k]; a.y = U[k + 1];
        v2f b; b.x = I[k]; b.y = I[k + 1];
        c = __builtin_amdgcn_wmma_f32_16x16x4_f32(
                /*neg_a=*/false, a, /*neg_b=*/false, b,
                /*c_mod=*/(short)0, c, /*reuse_a=*/false, /*reuse_b=*/false);
    }

    // Extract diagonal element owned by this lane (if any) and scale by 1/16
    int sel = -1;
    if (lane < 8u)        sel = (int)lane;        // (m,m), m = lane      in 0..7
    else if (lane >= 24u) sel = (int)lane - 24;   // (m,m), m = lane - 16 in 8..15
    float cv = 0.f;
#pragma unroll
    for (int j = 0; j < 8; ++j)
        if (sel == j) cv = c[j];
    if (lane < 8u)
        out[wave * 16u + lane] = cv * 0.0625f;
    else if (lane >= 24u)
        out[wave * 16u + (lane - 16u)] = cv * 0.0625f;
}

// ---------------------------------------------------------------------------
extern "C" void kernel_launch(void* const* d_in, const int* in_sizes, int n_in,
                              void* d_out, int out_size, void* d_ws, size_t ws_size,
                              hipStream_t stream) {
    (void)in_sizes; (void)n_in; (void)out_size; (void)ws_size;

    const float* user_emb = (const float*)d_in[0];   // 100000 x 64
    const float* item_emb = (const float*)d_in[1];   //  50000 x 64
    const float* vals     = (const float*)d_in[2];   // 4M
    const int*   rows     = (const int*)  d_in[3];   // 4M
    const int*   cols     = (const int*)  d_in[4];   // 4M
    const int*   users    = (const int*)  d_in[5];   // 16384
    const int*   items    = (const int*)  d_in[6];   // 16384
    float*       out      = (float*)d_out;           // 16384

    const size_t SZ = (size_t)N_NODES * LATENT;      // 9.6M floats
    float* acc = (float*)d_ws;
    float* x   = acc + SZ;
    float* y   = x + SZ;

    const int ELEM_BLOCKS = 1200;                    // grid-stride over 2.4M float4
    lgcn_init<<<ELEM_BLOCKS, 256, 0, stream>>>((const float4*)user_emb,
                                               (const float4*)item_emb,
                                               (float4*)acc, (float4*)x);

    const int SPMM_BLOCKS = (N_EDGES * 32) / 256;    // 500000 blocks, 1 edge/wave
    for (int layer = 0; layer < N_LAYERS; ++layer) {
        lgcn_zero<<<ELEM_BLOCKS, 256, 0, stream>>>((float4*)y);
        lgcn_spmm<<<SPMM_BLOCKS, 256, 0, stream>>>(vals, rows, cols, x, y);
        lgcn_accadd<<<ELEM_BLOCKS, 256, 0, stream>>>((float4*)acc, (const float4*)y);
        float* t = x; x = y; y = t;
    }

    // exactly 1024 waves: 256 blocks x 128 threads (4 waves/block)
    lgcn_gamma_wmma<<<256, 128, 0, stream>>>(acc, users, items, out);
}